// IntConv_39058432589925
// MI455X (gfx1250) — compile-verified
//
#include <hip/hip_runtime.h>

typedef __attribute__((ext_vector_type(2))) float v2f;
typedef __attribute__((ext_vector_type(8))) float v8f;

#define NEG_SLOPE 0.1f
#define WC_STRIDE 68   // 68 % 64 == 4 -> A-frag row reads spread over all 64 banks
#define XS_STRIDE 136  // 136 % 64 == 8 -> B-frag half-wave rows hit disjoint bank groups

__device__ __forceinline__ float leaky_f(float x) {
    return x > 0.0f ? x : NEG_SLOPE * x;
}

// ---------------------------------------------------------------------------
// hyper1: hdn[b][f] = leaky(sum_k emb[b][k] * W1[f][k]);  b<32, f<512
// Map b = tid&31 so each wave shares one W1 row (broadcast-friendly).
// ---------------------------------------------------------------------------
__global__ void hyper1_kernel(const float* __restrict__ emb,
                              const float* __restrict__ W1,
                              float* __restrict__ hdn) {
    int t = blockIdx.x * blockDim.x + threadIdx.x;  // 32*512 threads
    int b = t & 31;
    int f = t >> 5;
    const float* e = emb + b * 512;
    const float* w = W1 + f * 512;
    float s = 0.0f;
#pragma unroll 8
    for (int k = 0; k < 512; ++k) s = fmaf(e[k], w[k], s);
    hdn[b * 512 + f] = leaky_f(s);
}

// ---------------------------------------------------------------------------
// hyper2: kern[b][j] = sum_f hdn[b][f] * W2[j][f];  j < 576 (= 64 ch * 9 taps)
// ---------------------------------------------------------------------------
__global__ void hyper2_kernel(const float* __restrict__ hdn,
                              const float* __restrict__ W2,
                              float* __restrict__ kern) {
    int t = blockIdx.x * blockDim.x + threadIdx.x;  // 32*576 threads
    int b = t & 31;
    int j = t >> 5;
    const float* hv = hdn + b * 512;
    const float* w  = W2 + j * 512;
    float s = 0.0f;
#pragma unroll 8
    for (int k = 0; k < 512; ++k) s = fmaf(hv[k], w[k], s);
    kern[b * 576 + j] = s;
}

// ---------------------------------------------------------------------------
// Fused: per-sample depthwise 3x3 conv -> leaky -> 1x1 conv (WMMA f32 GEMM)
// Block = (batch b, image row y): 512 threads = 16 waves.
//   Phase 1: stage Wc / bias / this-sample kernels in LDS.
//   Phase 2: depthwise conv + leaky for 64 ch x 128 px into LDS (xs).
//   Phase 3: D[64 x 128] = Wc[64x64] * xs[64x128] + bc via v_wmma_f32_16x16x4_f32.
// ---------------------------------------------------------------------------
__global__ __launch_bounds__(512, 2)
void fused_dw_pw_kernel(const float* __restrict__ xin_g,
                        const float* __restrict__ kern,
                        const float* __restrict__ Wc,
                        const float* __restrict__ bc,
                        float* __restrict__ out) {
    __shared__ float wc_s[64 * WC_STRIDE];
    __shared__ float xs[64 * XS_STRIDE];
    __shared__ float kern_s[64 * 9];
    __shared__ float bc_s[64];

    const int tid = threadIdx.x;
    const int y = blockIdx.x & 127;
    const int b = blockIdx.x >> 7;

    // ---- Phase 1: stage weights ----
    for (int i = tid; i < 64 * 64; i += 512) {
        int o = i >> 6, c = i & 63;
        wc_s[o * WC_STRIDE + c] = Wc[i];
    }
    for (int i = tid; i < 576; i += 512) kern_s[i] = kern[b * 576 + i];
    if (tid < 64) bc_s[tid] = bc[tid];
    __syncthreads();

    // ---- Phase 2: depthwise 3x3 (cross-correlation, zero pad) + leaky ----
    const float* xin = xin_g + (size_t)(b * 64) * 16384;
#pragma unroll 4
    for (int j = 0; j < 16; ++j) {
        int i  = tid + j * 512;   // 0..8191 == c*128 + px
        int c  = i >> 7;
        int px = i & 127;
        const float* kr  = kern_s + c * 9;
        const float* row = xin + c * 16384 + y * 128 + px;
        float s = 0.0f;
#pragma unroll
        for (int ky = -1; ky <= 1; ++ky) {
            int yy = y + ky;
            if (yy < 0 || yy > 127) continue;
            const float* r = row + ky * 128;
#pragma unroll
            for (int kx = -1; kx <= 1; ++kx) {
                int xx = px + kx;
                if (xx < 0 || xx > 127) continue;
                s = fmaf(r[kx], kr[(ky + 1) * 3 + (kx + 1)], s);
            }
        }
        xs[c * XS_STRIDE + px] = leaky_f(s);
    }
    __syncthreads();

    // ---- Phase 3: WMMA GEMM D = Wc * xs ----
    const int lane = tid & 31;
    const int wave = tid >> 5;
    const int half = lane >> 4;   // 0: lanes 0-15, 1: lanes 16-31
    const int ln   = lane & 15;
    const int m0 = (wave >> 2) * 16;          // out-channel tile: 0,16,32,48
    const int n0 = (wave & 3) * 16;           // pixel tiles n0 and n0+64

    v8f acc0 = {};
    v8f acc1 = {};
#pragma unroll
    for (int kt = 0; kt < 16; ++kt) {
        const int k0 = kt * 4;
        // A frag (16x4 of Wc): lane<16 -> K{k0,k0+1}, lane>=16 -> K{k0+2,k0+3}
        const float* ap = wc_s + (m0 + ln) * WC_STRIDE + k0 + 2 * half;
        v2f a;
        a[0] = ap[0];
        a[1] = ap[1];
        // B frag (4x16 of xs), same K split across half-waves
        const float* bp0 = xs + (k0 + 2 * half) * XS_STRIDE + n0 + ln;
        v2f b0;
        b0[0] = bp0[0];
        b0[1] = bp0[XS_STRIDE];
        acc0 = __builtin_amdgcn_wmma_f32_16x16x4_f32(
            false, a, false, b0, (short)0, acc0, false, false);
        const float* bp1 = bp0 + 64;
        v2f b1;
        b1[0] = bp1[0];
        b1[1] = bp1[XS_STRIDE];
        acc1 = __builtin_amdgcn_wmma_f32_16x16x4_f32(
            false, a, false, b1, (short)0, acc1, false, false);
    }

    // ---- Store: C/D layout — VGPR v holds M = m0 + v + 8*half, N = n + n0 ----
    float* outp = out + (size_t)(b * 64) * 16384 + y * 128;
#pragma unroll
    for (int v = 0; v < 8; ++v) {
        const int m = m0 + v + 8 * half;
        const float bias = bc_s[m];
        outp[m * 16384 + n0 + ln]      = acc0[v] + bias;
        outp[m * 16384 + n0 + 64 + ln] = acc1[v] + bias;
    }
}

// ---------------------------------------------------------------------------
extern "C" void kernel_launch(void* const* d_in, const int* in_sizes, int n_in,
                              void* d_out, int out_size, void* d_ws, size_t ws_size,
                              hipStream_t stream) {
    const float* x   = (const float*)d_in[0];  // [32,64,128,128]
    const float* emb = (const float*)d_in[1];  // [32,512]
    const float* W1  = (const float*)d_in[2];  // [512,512]
    const float* W2  = (const float*)d_in[3];  // [576,512]
    const float* Wc  = (const float*)d_in[4];  // [64,64]
    const float* bc  = (const float*)d_in[5];  // [64]
    float* out = (float*)d_out;

    float* hdn  = (float*)d_ws;          // 32*512 f32
    float* kern = hdn + 32 * 512;        // 32*576 f32

    hyper1_kernel<<<(32 * 512) / 256, 256, 0, stream>>>(emb, W1, hdn);
    hyper2_kernel<<<(32 * 576) / 256, 256, 0, stream>>>(hdn, W2, kern);
    fused_dw_pw_kernel<<<32 * 128, 512, 0, stream>>>(x, kern, Wc, bc, out);
}